// SLP_GIN_4node_52871047413799
// MI455X (gfx1250) — compile-verified
//
#include <hip/hip_runtime.h>
#include <hip/hip_bf16.h>

typedef float v2f __attribute__((ext_vector_type(2)));
typedef float v8f __attribute__((ext_vector_type(8)));

#define N_NODES   50000
#define N_EDGES   800000
#define FEATS     128

// ---------------------------------------------------------------------------
// Utility: zero a float buffer (grid-stride)
// ---------------------------------------------------------------------------
__global__ void zero_f32_kernel(float* __restrict__ p, size_t n) {
    size_t i = (size_t)blockIdx.x * blockDim.x + threadIdx.x;
    size_t stride = (size_t)gridDim.x * blockDim.x;
    for (; i < n; i += stride) p[i] = 0.0f;
}

// ---------------------------------------------------------------------------
// Degree count: deg[dst[e]] += 1
// ---------------------------------------------------------------------------
__global__ void deg_count_kernel(const int* __restrict__ dst, float* __restrict__ deg,
                                 int n_edges) {
    int e = blockIdx.x * blockDim.x + threadIdx.x;
    if (e < n_edges) atomicAdd(&deg[dst[e]], 1.0f);
}

// invdeg[i] = 1 / max(deg[i], 1)
__global__ void deg_invert_kernel(float* __restrict__ deg, int n) {
    int i = blockIdx.x * blockDim.x + threadIdx.x;
    if (i < n) deg[i] = 1.0f / fmaxf(deg[i], 1.0f);
}

// ---------------------------------------------------------------------------
// Edge scatter: AGG[dst] += H[src]   (32 threads per edge, 4 floats each)
// H, AGG are [N_NODES x 128]; both fit in the 192MB L2 so atomics stay on-chip
// ---------------------------------------------------------------------------
__global__ __launch_bounds__(256)
void gin_scatter_kernel(const float* __restrict__ H, const int* __restrict__ src,
                        const int* __restrict__ dst, float* __restrict__ AGG,
                        int n_edges) {
    long long tid = (long long)blockIdx.x * blockDim.x + threadIdx.x;
    int e = (int)(tid >> 5);
    if (e >= n_edges) return;
    int f = (int)(tid & 31) * 4;
    int s = src[e];
    int d = dst[e];
    const float4 hv = *(const float4*)(H + (size_t)s * FEATS + f);
    float* o = AGG + (size_t)d * FEATS + f;
    atomicAdd(o + 0, hv.x);
    atomicAdd(o + 1, hv.y);
    atomicAdd(o + 2, hv.z);
    atomicAdd(o + 3, hv.w);
}

// ---------------------------------------------------------------------------
// Fused GIN-layer GEMM with f32 WMMA (16x16x4):
//   A[m,k] = COMBINE ? (1+eps[eidx])*H[m,k] + invdeg[m]*AGG[m,k] : H[m,k]
//   OUT[M x N] = act( A @ W^T + bias ),  W is [N x K], K = 128
// One wave owns a 16-row strip and NTILES 16x16 accumulator tiles.
//
// f32 WMMA fragment layouts (ISA 7.12.2):
//   A 16x4: lane = m%16 + 16*khalf, VGPRv -> K = 2*khalf + v
//   B 4x16: lane = n%16 + 16*khalf, VGPRv -> K = 2*khalf + v   (B[k,n] = W[n,k])
//   C 16x16: VGPRv, lane -> row = v + 8*(lane>=16), col = lane%16
// ---------------------------------------------------------------------------
template <int NTILES, bool RELU, bool COMBINE>
__global__ __launch_bounds__(256)
void gin_gemm_wmma_kernel(const float* __restrict__ H, const float* __restrict__ AGG,
                          const float* __restrict__ invdeg,
                          const float* __restrict__ eps, int eps_idx,
                          const float* __restrict__ W, const float* __restrict__ bias,
                          float* __restrict__ OUT, int M) {
    constexpr int K = FEATS;
    constexpr int N = NTILES * 16;

    const int wave = (int)((blockIdx.x * blockDim.x + threadIdx.x) >> 5);
    const int lane = threadIdx.x & 31;
    const int m0 = wave * 16;
    if (m0 >= M) return;                 // wave-uniform: EXEC stays all-1s for WMMA

    const int mlo   = lane & 15;
    const int khalf = lane >> 4;         // 0 or 1
    const int mrow  = m0 + mlo;

    float e1 = 1.0f, invd = 0.0f;
    if (COMBINE) {
        e1   = 1.0f + eps[eps_idx];
        invd = invdeg[mrow];
    }

    v8f acc[NTILES];
#pragma unroll
    for (int t = 0; t < NTILES; ++t) acc[t] = v8f{};

    const float* __restrict__ arow = H + (size_t)mrow * K;
    const float* __restrict__ grow = COMBINE ? (AGG + (size_t)mrow * K) : nullptr;

    for (int k0 = 0; k0 < K; k0 += 4) {
        const int ka = k0 + 2 * khalf;
        v2f a = *(const v2f*)(arow + ka);
        if (COMBINE) {
            v2f g = *(const v2f*)(grow + ka);
            a.x = e1 * a.x + invd * g.x;
            a.y = e1 * a.y + invd * g.y;
        }
#pragma unroll
        for (int t = 0; t < NTILES; ++t) {
            // B[k, n] = W[n, k]; per-lane contiguous float2 along K
            const float* wrow = W + (size_t)(t * 16 + mlo) * K + ka;
            v2f bf = *(const v2f*)wrow;
            acc[t] = __builtin_amdgcn_wmma_f32_16x16x4_f32(
                /*neg_a=*/false, a, /*neg_b=*/false, bf,
                /*c_mod=*/(short)0, acc[t], /*reuse_a=*/false, /*reuse_b=*/false);
        }
    }

    // Epilogue: bias (+ ReLU) and store
#pragma unroll
    for (int t = 0; t < NTILES; ++t) {
        const int n  = t * 16 + mlo;
        const float bv = bias[n];
#pragma unroll
        for (int v = 0; v < 8; ++v) {
            const int row = m0 + v + 8 * khalf;
            float val = acc[t][v] + bv;
            if (RELU) val = fmaxf(val, 0.0f);
            OUT[(size_t)row * N + n] = val;
        }
    }
}

// ---------------------------------------------------------------------------
// Host-side orchestration
// ---------------------------------------------------------------------------
static inline int cdiv(long long a, long long b) { return (int)((a + b - 1) / b); }

extern "C" void kernel_launch(void* const* d_in, const int* in_sizes, int n_in,
                              void* d_out, int out_size, void* d_ws, size_t ws_size,
                              hipStream_t stream) {
    const float* inputs = (const float*)d_in[0];   // [50000,128]
    const float* fc_W   = (const float*)d_in[1];   // [128,128]
    const float* fc_b   = (const float*)d_in[2];   // [128]
    const float* W      = (const float*)d_in[3];   // [3,128,128]
    const float* b      = (const float*)d_in[4];   // [3,128]
    const float* W_last = (const float*)d_in[5];   // [64,128]
    const float* b_last = (const float*)d_in[6];   // [64]
    const float* eps    = (const float*)d_in[7];   // [4]
    const int*   src    = (const int*)d_in[8];     // [800000]
    const int*   dst    = (const int*)d_in[9];     // [800000]
    float* out = (float*)d_out;                    // [50000,64]

    const size_t HF = (size_t)N_NODES * FEATS;     // 6.4M floats = 25.6MB
    float* h_a    = (float*)d_ws;
    float* h_b    = h_a + HF;
    float* agg    = h_b + HF;
    float* invdeg = agg + HF;                      // N_NODES floats
    (void)ws_size; (void)in_sizes; (void)n_in; (void)out_size;

    const int mwaves  = N_NODES / 16;              // 3125 waves (M % 16 == 0)
    const int gblocks = cdiv(mwaves, 8);           // 8 waves / 256-thread block
    const int sblocks = cdiv((long long)N_EDGES * 32, 256);

    // ---- degrees (computed once) ----
    zero_f32_kernel<<<256, 256, 0, stream>>>(invdeg, (size_t)N_NODES);
    deg_count_kernel<<<cdiv(N_EDGES, 256), 256, 0, stream>>>(dst, invdeg, N_EDGES);
    deg_invert_kernel<<<cdiv(N_NODES, 256), 256, 0, stream>>>(invdeg, N_NODES);

    // ---- input SLP: h = relu(inputs @ fc_W^T + fc_b) ----
    gin_gemm_wmma_kernel<8, true, false><<<gblocks, 256, 0, stream>>>(
        inputs, nullptr, nullptr, eps, 0, fc_W, fc_b, h_a, N_NODES);

    // ---- 3 hidden GIN layers ----
    float* cur = h_a;
    float* nxt = h_b;
    for (int i = 0; i < 3; ++i) {
        zero_f32_kernel<<<1024, 256, 0, stream>>>(agg, HF);
        gin_scatter_kernel<<<sblocks, 256, 0, stream>>>(cur, src, dst, agg, N_EDGES);
        gin_gemm_wmma_kernel<8, true, true><<<gblocks, 256, 0, stream>>>(
            cur, agg, invdeg, eps, i, W + (size_t)i * FEATS * FEATS, b + (size_t)i * FEATS,
            nxt, N_NODES);
        float* tmp = cur; cur = nxt; nxt = tmp;
    }

    // ---- last GIN layer (N=64, no ReLU) ----
    zero_f32_kernel<<<1024, 256, 0, stream>>>(agg, HF);
    gin_scatter_kernel<<<sblocks, 256, 0, stream>>>(cur, src, dst, agg, N_EDGES);
    gin_gemm_wmma_kernel<4, false, true><<<gblocks, 256, 0, stream>>>(
        cur, agg, invdeg, eps, 3, W_last, b_last, out, N_NODES);
}